// Net_74826920231498
// MI455X (gfx1250) — compile-verified
//
#include <hip/hip_runtime.h>
#include <stdint.h>

typedef __attribute__((ext_vector_type(2))) float v2f;
typedef __attribute__((ext_vector_type(8))) float v8f;
typedef __attribute__((ext_vector_type(4))) unsigned int v4u;
typedef __attribute__((ext_vector_type(8))) int v8i;
typedef __attribute__((ext_vector_type(4))) int v4i;

// ---------------------------------------------------------------------------
// TDM: DMA a tile0 x tile1 (elements) f32 tile from global (row pitch
// pitchElems) into LDS at ldsDst (contiguous, row length == tile0).
// Descriptor packed per CDNA5 ISA.
// tensor_dim == tile_dim (caller pre-clips; no OOB semantics relied upon).
// ---------------------------------------------------------------------------
__device__ inline void tdm_load_2d(float* ldsDst, const float* gsrc,
                                   int tile0, int tile1, long long pitchElems) {
  unsigned lds_addr = (unsigned)(size_t)(void*)ldsDst;          // low 32 bits of flat = LDS byte offset
  unsigned long long ga = (unsigned long long)(size_t)(const void*)gsrc;
  unsigned long long q;
  v4u g0; v8i g1; v4i g2, g3; v8i g4;

  // group0: [1:0] count=1 ; [63:32] lds_addr ; [120:64] global_addr ; [127:126] type=2
  q = 1ull | ((unsigned long long)lds_addr << 32);
  g0[0] = (unsigned)q; g0[1] = (unsigned)(q >> 32);
  q = (ga & 0x01FFFFFFFFFFFFFFull) | (2ull << 62);
  g0[2] = (unsigned)q; g0[3] = (unsigned)(q >> 32);

  unsigned long long td0 = (unsigned)tile0;
  unsigned long long td1 = (unsigned)tile1;
  unsigned long long st0 = (unsigned long long)pitchElems & 0xFFFFFFFFFFFFull;
  // group1 bits: [15:0] wg_mask=0 [17:16] data_size=2(4B) [79:48] tensor_dim0
  // [111:80] tensor_dim1 [127:112] tile_dim0 [143:128] tile_dim1 [159:144] tile_dim2=0
  // [207:160] tensor_dim0_stride [255:208] tensor_dim1_stride=0
  q = (2ull << 16) | (td0 << 48);
  g1[0] = (int)q; g1[1] = (int)(q >> 32);
  q = (td0 >> 16) | (td1 << 16) | ((td0 & 0xFFFFull) << 48);
  g1[2] = (int)q; g1[3] = (int)(q >> 32);
  q = (td1 & 0xFFFFull) | (st0 << 32);
  g1[4] = (int)q; g1[5] = (int)(q >> 32);
  q = (st0 >> 32) & 0xFFFFull;
  g1[6] = (int)q; g1[7] = (int)(q >> 32);

  // group2: tensor_dim2=1, tensor_dim3=1, tensor_dim2_stride=0, tile_dim3=0
  g2[0] = 1; g2[1] = 1; g2[2] = 0; g2[3] = 0;
  // group3: tensor_dim3_stride=0, tensor_dim4=1 (bits 79:48), tile_dim4=0
  g3[0] = 0; g3[1] = 0x00010000; g3[2] = 0; g3[3] = 0;
  // group4 (6-arg builtin form): unused, zero
  g4[0] = 0; g4[1] = 0; g4[2] = 0; g4[3] = 0;
  g4[4] = 0; g4[5] = 0; g4[6] = 0; g4[7] = 0;

  __builtin_amdgcn_tensor_load_to_lds(g0, g1, g2, g3, g4, 0);
}

// ---------------------------------------------------------------------------
// Plain VALU 3x3 conv over an LDS tile (used for non-16-aligned halo layers).
// dst may have a padded pitch (DSTP >= DSTW) for downstream WMMA.
// ---------------------------------------------------------------------------
template <int SRCW, int DSTW, int DSTH, int DSTP>
__device__ inline void conv3x3_valu(const float* __restrict__ src,
                                    float* __restrict__ dst,
                                    const float* wv, int tid) {
  for (int idx = tid; idx < DSTW * DSTH; idx += 256) {
    int r = idx / DSTW;
    int c = idx - r * DSTW;
    const float* p = src + r * SRCW + c;
    float acc = p[0] * wv[0];
    acc = fmaf(p[1], wv[1], acc);
    acc = fmaf(p[2], wv[2], acc);
    acc = fmaf(p[SRCW + 0], wv[3], acc);
    acc = fmaf(p[SRCW + 1], wv[4], acc);
    acc = fmaf(p[SRCW + 2], wv[5], acc);
    acc = fmaf(p[2 * SRCW + 0], wv[6], acc);
    acc = fmaf(p[2 * SRCW + 1], wv[7], acc);
    acc = fmaf(p[2 * SRCW + 2], wv[8], acc);
    dst[r * DSTP + c] = acc;
  }
}

// ---------------------------------------------------------------------------
// 3x3 conv producing a 64x64 tile via V_WMMA_F32_16X16X4_F32.
// D(16x16) = sum_{dr=0..2} A_dr(16x20) * B_dr(20x16), B = banded Toeplitz:
// B[k,n] = w[dr][k-n] for 0<=k-n<=2 else 0 (k=18,19 are zero padding).
// src is a 66-row tile with pitch 68; columns 66,67 of every row are zeroed by
// the caller so the k=18/19 loads are unconditional, in-bounds, NaN-free, and
// the lane->bank mapping (row step 68 words = 4 banks) is conflict-free.
// A layout (16x4 f32): lanes 0-15 hold M=lane, K={0,1}; lanes 16-31 M=lane-16, K={2,3}.
// B layout (4x16 f32): VGPR0 = rows K=0 / K=2 ; VGPR1 = K=1 / K=3.
// D layout: VGPR j -> M = j (+8 for upper lanes), N = lane&15.
// ---------------------------------------------------------------------------
__device__ inline void conv3x3_wmma64(const float* __restrict__ src,  // pitch 68
                                      float* dst, int dstPitch,
                                      const float* wv, int tid) {
  const int lane = tid & 31;
  const int wave = tid >> 5;
  const int n = lane & 15;
  const int hi = lane >> 4;            // 0 or 1
  for (int blk = wave; blk < 16; blk += 8) {
    const int r0 = (blk >> 2) * 16;
    const int c0 = (blk & 3) * 16;
    v8f acc = {0.f, 0.f, 0.f, 0.f, 0.f, 0.f, 0.f, 0.f};
#pragma unroll
    for (int dr = 0; dr < 3; ++dr) {
      const float w0 = wv[dr * 3 + 0];
      const float w1 = wv[dr * 3 + 1];
      const float w2 = wv[dr * 3 + 2];
#pragma unroll
      for (int k0 = 0; k0 < 20; k0 += 4) {
        const int kl = k0 + (hi << 1);            // even; this lane's first K
        const float* sp = src + (r0 + dr + n) * 68 + (c0 + kl);
        v2f a;
        a.x = sp[0];
        a.y = sp[1];
        const int d0 = kl - n;
        const int d1 = d0 + 1;
        v2f b;
        b.x = (d0 == 0) ? w0 : (d0 == 1) ? w1 : (d0 == 2) ? w2 : 0.f;
        b.y = (d1 == 0) ? w0 : (d1 == 1) ? w1 : (d1 == 2) ? w2 : 0.f;
        acc = __builtin_amdgcn_wmma_f32_16x16x4_f32(
            false, a, false, b, (short)0, acc, false, false);
      }
    }
    float* dp = dst + (r0 + hi * 8) * dstPitch + c0 + n;
#pragma unroll
    for (int j = 0; j < 8; ++j) dp[j * dstPitch] = acc[j];
  }
}

// ---------------------------------------------------------------------------
// Kernel A: conv1 -> conv2 -> maxpool2x2, fused per 64x64 (full-res) tile.
// Input tile 68x68 (halo 2) TDM'd into LDS; pooled 32x32 written to d_ws.
// ---------------------------------------------------------------------------
__global__ __launch_bounds__(256) void convpool_front(const float* __restrict__ x,
                                                      const float* __restrict__ w1p,
                                                      const float* __restrict__ w2p,
                                                      float* __restrict__ P) {
  __shared__ float smem[4624 + 4488 + 4096];   // 68x68 + 66rows*pitch68 + 64x64
  float* inT = smem;           // 68 x 68
  float* t1  = smem + 4624;    // 66 rows, pitch 68 (cols 66,67 zeroed)
  float* t2  = t1 + 4488;      // 64 x 64
  const int tid = threadIdx.x;
  const int bx = blockIdx.x, by = blockIdx.y;
  const int gx0 = bx * 64 - 2, gy0 = by * 64 - 2;

  // zero the 2 pad columns of t1 (read by k=18/19 with zero B weights)
  if (tid < 66) { t1[tid * 68 + 66] = 0.f; t1[tid * 68 + 67] = 0.f; }

  const bool clipped = (gx0 < 0) | (gy0 < 0) | (gx0 + 68 > 8192) | (gy0 + 68 > 8192);
  if (clipped) {
    for (int i = tid; i < 4624; i += 256) inT[i] = 0.f;
  }

  float wA[9], wB[9];
#pragma unroll
  for (int i = 0; i < 9; ++i) { wA[i] = w1p[i]; wB[i] = w2p[i]; }

  __syncthreads();

  if (tid < 32) {   // one wave issues the TDM DMA
    const int offX = gx0 < 0 ? -gx0 : 0;
    const int offY = gy0 < 0 ? -gy0 : 0;
    const int sx = gx0 + offX, sy = gy0 + offY;
    const int t0 = min(68 - offX, 8192 - sx);
    const int t1d = min(68 - offY, 8192 - sy);
    tdm_load_2d(inT + offY * 68 + offX, x + (long long)sy * 8192 + sx, t0, t1d, 8192);
    __builtin_amdgcn_s_wait_tensorcnt(0);
  }
  __syncthreads();

  conv3x3_valu<68, 66, 66, 68>(inT, t1, wA, tid);
  __syncthreads();
  conv3x3_wmma64(t1, t2, 64, wB, tid);
  __syncthreads();

  // 2x2 maxpool of t2 -> pooled image in d_ws
  for (int i = tid; i < 1024; i += 256) {
    const int pr = i >> 5, pc = i & 31;
    const float* q = t2 + (pr * 2) * 64 + pc * 2;
    const float m = fmaxf(fmaxf(q[0], q[1]), fmaxf(q[64], q[65]));
    P[(long long)(by * 32 + pr) * 4096 + (bx * 32 + pc)] = m;
  }
}

// ---------------------------------------------------------------------------
// Kernel B: conv3 -> conv4 -> conv5 on the pooled 4096^2 image, fused per
// 64x64 output tile. Input tile 70x70 (halo 3) via TDM; conv5 via WMMA
// written straight to global output.
// ---------------------------------------------------------------------------
__global__ __launch_bounds__(256) void conv_back(const float* __restrict__ P,
                                                 const float* __restrict__ w3p,
                                                 const float* __restrict__ w4p,
                                                 const float* __restrict__ w5p,
                                                 float* __restrict__ out) {
  __shared__ float smem[4900 + 4624 + 4488];   // 70x70 + 68x68 + 66rows*pitch68
  float* s0 = smem;            // 70 x 70
  float* s1 = smem + 4900;     // 68 x 68
  float* s2 = s1 + 4624;       // 66 rows, pitch 68 (cols 66,67 zeroed)
  const int tid = threadIdx.x;
  const int bx = blockIdx.x, by = blockIdx.y;
  const int gx0 = bx * 64 - 3, gy0 = by * 64 - 3;

  if (tid < 66) { s2[tid * 68 + 66] = 0.f; s2[tid * 68 + 67] = 0.f; }

  const bool clipped = (gx0 < 0) | (gy0 < 0) | (gx0 + 70 > 4096) | (gy0 + 70 > 4096);
  if (clipped) {
    for (int i = tid; i < 4900; i += 256) s0[i] = 0.f;
  }

  float wC[9], wD[9], wE[9];
#pragma unroll
  for (int i = 0; i < 9; ++i) { wC[i] = w3p[i]; wD[i] = w4p[i]; wE[i] = w5p[i]; }

  __syncthreads();

  if (tid < 32) {
    const int offX = gx0 < 0 ? -gx0 : 0;
    const int offY = gy0 < 0 ? -gy0 : 0;
    const int sx = gx0 + offX, sy = gy0 + offY;
    const int t0 = min(70 - offX, 4096 - sx);
    const int t1d = min(70 - offY, 4096 - sy);
    tdm_load_2d(s0 + offY * 70 + offX, P + (long long)sy * 4096 + sx, t0, t1d, 4096);
    __builtin_amdgcn_s_wait_tensorcnt(0);
  }
  __syncthreads();

  conv3x3_valu<70, 68, 68, 68>(s0, s1, wC, tid);
  __syncthreads();
  conv3x3_valu<68, 66, 66, 68>(s1, s2, wD, tid);
  __syncthreads();
  conv3x3_wmma64(s2, out + (long long)(by * 64) * 4096 + bx * 64, 4096, wE, tid);
}

// ---------------------------------------------------------------------------
extern "C" void kernel_launch(void* const* d_in, const int* in_sizes, int n_in,
                              void* d_out, int out_size, void* d_ws, size_t ws_size,
                              hipStream_t stream) {
  const float* x  = (const float*)d_in[0];
  const float* w1 = (const float*)d_in[1];
  const float* w2 = (const float*)d_in[2];
  const float* w3 = (const float*)d_in[3];
  const float* w4 = (const float*)d_in[4];
  const float* w5 = (const float*)d_in[5];
  float* out = (float*)d_out;
  float* P   = (float*)d_ws;   // pooled intermediate, 4096*4096 f32 = 64 MB

  dim3 blk(256);
  convpool_front<<<dim3(128, 128), blk, 0, stream>>>(x, w1, w2, P);
  conv_back<<<dim3(64, 64), blk, 0, stream>>>(P, w3, w4, w5, out);
}